// EdgePredictorGNN_43430709297956
// MI455X (gfx1250) — compile-verified
//
#include <hip/hip_runtime.h>
#include <hip/hip_bf16.h>

typedef __attribute__((ext_vector_type(2))) float v2f;
typedef __attribute__((ext_vector_type(8))) float v8f;

#define CH 128  // IN_CH == HID_CH == 128

// ---------------------------------------------------------------------------
// Degree / normalization kernels
// ---------------------------------------------------------------------------
__global__ __launch_bounds__(256) void gcn_deg_init(float* deg, int n) {
  int i = blockIdx.x * 256 + threadIdx.x;
  if (i < n) deg[i] = 1.0f;  // self-loop contributes 1 to every node's degree
}

__global__ __launch_bounds__(256) void gcn_deg_count(const long long* __restrict__ dst,
                                                     float* __restrict__ deg, int E) {
  int e = blockIdx.x * 256 + threadIdx.x;
  if (e < E) atomicAdd(&deg[(int)dst[e]], 1.0f);
}

__global__ __launch_bounds__(256) void gcn_rsqrt(float* d, int n) {
  int i = blockIdx.x * 256 + threadIdx.x;
  if (i < n) d[i] = rsqrtf(d[i]);  // deg >= 1 always (self-loops)
}

// ---------------------------------------------------------------------------
// f32 WMMA GEMM: C[M x 128] = A[M x 128] * B[128 x 128]   (all row-major)
// One 256-thread block = 8 waves; block owns rows [16*bx, 16*bx+16),
// wave w owns columns [16*w, 16*w+16). K consumed 4 at a time by
// V_WMMA_F32_16X16X4_F32.
// ---------------------------------------------------------------------------
__global__ __launch_bounds__(256) void gcn_gemm_f32(const float* __restrict__ A,
                                                    const float* __restrict__ B,
                                                    float* __restrict__ C, int M) {
  const int lane = threadIdx.x & 31;
  const int wave = threadIdx.x >> 5;
  const int m0 = blockIdx.x * 16;
  const int n0 = wave * 16;
  const int half = lane >> 4;   // 0: K pair {0,1}; 1: K pair {2,3}
  const int l = lane & 15;

  // A fragment source: row (m0+l), k-offset half*2 within each K-step of 4
  const float* arow = A + (size_t)(m0 + l) * CH + half * 2;
  // B fragment source: col (n0+l), rows half*2 and half*2+1 within each K-step
  const float* bcol = B + (size_t)(half * 2) * CH + n0 + l;

  v8f c = {};
#pragma unroll
  for (int kk = 0; kk < CH; kk += 4) {
    v2f a, b;
    a.x = arow[kk + 0];
    a.y = arow[kk + 1];
    b.x = bcol[(size_t)kk * CH];
    b.y = bcol[(size_t)kk * CH + CH];
    // (neg_a, A, neg_b, B, c_mod, C, reuse_a, reuse_b)
    c = __builtin_amdgcn_wmma_f32_16x16x4_f32(false, a, false, b, (short)0, c,
                                              false, false);
  }

  // C layout: VGPR i -> row m0 + i + half*8, col n0 + l
  float* crow = C + (size_t)(m0 + half * 8) * CH + n0 + l;
#pragma unroll
  for (int i = 0; i < 8; ++i) crow[(size_t)i * CH] = c[i];
}

// ---------------------------------------------------------------------------
// Self-loop init: out[i][c] = dinv[i]^2 * h[i][c]   (seeds the accumulator)
// ---------------------------------------------------------------------------
__global__ __launch_bounds__(256) void gcn_self_init(const float* __restrict__ dinv,
                                                     const float* __restrict__ h,
                                                     float* __restrict__ out, int n) {
  long long i = (long long)blockIdx.x * 256 + threadIdx.x;
  long long total = (long long)n * CH;
  if (i >= total) return;
  int node = (int)(i >> 7);  // CH == 128
  float di = dinv[node];
  out[i] = di * di * h[i];
}

// ---------------------------------------------------------------------------
// Edge aggregation: out[dst] += dinv[s]*dinv[d] * h[src]
// One wave per edge: 32 lanes x float4 = 128 channels; f32 atomics hit L2.
// ---------------------------------------------------------------------------
__global__ __launch_bounds__(256) void gcn_edge_agg(const long long* __restrict__ src,
                                                    const long long* __restrict__ dst,
                                                    const float* __restrict__ dinv,
                                                    const float* __restrict__ h,
                                                    float* __restrict__ out, int E) {
  int e = blockIdx.x * 8 + (threadIdx.x >> 5);
  if (e >= E) return;
  int lane = threadIdx.x & 31;
  int s = (int)src[e];
  int d = (int)dst[e];
  float coef = dinv[s] * dinv[d];
  const float4* hp = (const float4*)(h + (size_t)s * CH);
  float4 v = hp[lane];
  float* op = out + (size_t)d * CH + lane * 4;
  atomicAdd(op + 0, coef * v.x);
  atomicAdd(op + 1, coef * v.y);
  atomicAdd(op + 2, coef * v.z);
  atomicAdd(op + 3, coef * v.w);
}

// ---------------------------------------------------------------------------
// Finalize: out = relu(out + bias)
// ---------------------------------------------------------------------------
__global__ __launch_bounds__(256) void gcn_bias_relu(float* __restrict__ out,
                                                     const float* __restrict__ b, int n) {
  long long i = (long long)blockIdx.x * 256 + threadIdx.x;
  long long total = (long long)n * CH;
  if (i >= total) return;
  float v = out[i] + b[(int)(i & (CH - 1))];
  out[i] = v > 0.0f ? v : 0.0f;
}

// ---------------------------------------------------------------------------
extern "C" void kernel_launch(void* const* d_in, const int* in_sizes, int n_in,
                              void* d_out, int out_size, void* d_ws, size_t ws_size,
                              hipStream_t stream) {
  const float* x      = (const float*)d_in[0];
  const long long* ei = (const long long*)d_in[1];  // jnp.int64, shape (2, E)
  const float* W1     = (const float*)d_in[2];
  const float* b1     = (const float*)d_in[3];
  const float* W2     = (const float*)d_in[4];
  const float* b2     = (const float*)d_in[5];
  float* out          = (float*)d_out;

  const int N = in_sizes[0] / CH;  // 50000
  const int E = in_sizes[1] / 2;   // 800000
  const long long* src = ei;
  const long long* dst = ei + E;

  float* ws   = (float*)d_ws;
  float* dinv = ws;                          // N floats
  float* h    = ws + N;                      // N*128 floats (GEMM out, both layers)
  float* a1   = h + (size_t)N * CH;          // N*128 floats (layer-1 aggregate)

  const int elemBlocks = (int)(((long long)N * CH + 255) / 256);
  const int edgeBlocks = (E + 7) / 8;

  // --- normalization coefficients ---
  gcn_deg_init<<<(N + 255) / 256, 256, 0, stream>>>(dinv, N);
  gcn_deg_count<<<(E + 255) / 256, 256, 0, stream>>>(dst, dinv, E);
  gcn_rsqrt<<<(N + 255) / 256, 256, 0, stream>>>(dinv, N);

  // --- layer 1 ---
  gcn_gemm_f32<<<N / 16, 256, 0, stream>>>(x, W1, h, N);
  gcn_self_init<<<elemBlocks, 256, 0, stream>>>(dinv, h, a1, N);
  gcn_edge_agg<<<edgeBlocks, 256, 0, stream>>>(src, dst, dinv, h, a1, E);
  gcn_bias_relu<<<elemBlocks, 256, 0, stream>>>(a1, b1, N);

  // --- layer 2 (aggregate straight into d_out) ---
  gcn_gemm_f32<<<N / 16, 256, 0, stream>>>(a1, W2, h, N);
  gcn_self_init<<<elemBlocks, 256, 0, stream>>>(dinv, h, out, N);
  gcn_edge_agg<<<edgeBlocks, 256, 0, stream>>>(src, dst, dinv, h, out, E);
  gcn_bias_relu<<<elemBlocks, 256, 0, stream>>>(out, b2, N);
}